// GraphGAT_72919954751488
// MI455X (gfx1250) — compile-verified
//
#include <hip/hip_runtime.h>
#include <hip/hip_bf16.h>
#include <math.h>

// ---------------------------------------------------------------------------
// Problem constants (match reference)
// ---------------------------------------------------------------------------
#define NN    50000
#define EE    800000
#define ET    (EE + NN)      // edges + self loops
#define DIN   128
#define DHC   128
#define DOUTC 64
#define NGP   128
#define EPSV  1e-5f
#define SLOPE 0.2f

typedef __attribute__((ext_vector_type(2))) float v2f;
typedef __attribute__((ext_vector_type(8))) float v8f;

// ---------------------------------------------------------------------------
// Utility kernels
// ---------------------------------------------------------------------------
__global__ void fill_f32(float* __restrict__ p, int n, float v) {
    int i = blockIdx.x * blockDim.x + threadIdx.x;
    if (i < n) p[i] = v;
}

__global__ void build_edges(const long long* __restrict__ ei,
                            int* __restrict__ srcI, int* __restrict__ dstI) {
    int i = blockIdx.x * blockDim.x + threadIdx.x;
    if (i >= ET) return;
    if (i < EE) {
        srcI[i] = (int)ei[i];          // row 0: src
        dstI[i] = (int)ei[EE + i];     // row 1: dst
    } else {
        srcI[i] = i - EE;              // self loops
        dstI[i] = i - EE;
    }
}

// ---------------------------------------------------------------------------
// Dense GEMM, f32, via V_WMMA_F32_16X16X4_F32 with 4-tile register blocking.
// One wave computes a 16x64 slab of C (4 accumulators): the A fragment is
// loaded once per k-step and reused by 4 back-to-back WMMAs (hides XDL
// latency, 4x WMMA density per VMEM op).
//  A: M x K (row major), B: K x Ncols (row major), C: M x Ncols. Ncols%64==0.
//  A frag (16x4 f32): lanes 0-15 M=lane,K={k,k+1}; lanes 16-31 M=lane-16,K={k+2,k+3}
//  B frag (4x16 f32): v0 lanes0-15 K=k row, lanes16-31 K=k+2 row; v1 K=k+1 / k+3
//  C/D  : VGPR i, lanes0-15 -> M=i,N=lane ; lanes16-31 -> M=i+8,N=lane-16
// ---------------------------------------------------------------------------
__global__ void gemm_wmma_f32(const float* __restrict__ A,
                              const float* __restrict__ B,
                              float* __restrict__ C,
                              int M, int K, int Ncols) {
    const int slabsN = Ncols >> 6;             // groups of 4 N-tiles
    const int total  = (M >> 4) * slabsN;
    int w = blockIdx.x * (blockDim.x >> 5) + (threadIdx.x >> 5);
    if (w >= total) return;
    const int tm   = w / slabsN;
    const int tn4  = w % slabsN;
    const int lane = threadIdx.x & 31;
    const int half = lane >> 4;   // 0 / 1
    const int r    = lane & 15;

    const float* Arow  = A + (size_t)(tm * 16 + r) * K;
    const float* Bbase = B + tn4 * 64 + r;

    v8f acc0 = {}, acc1 = {}, acc2 = {}, acc3 = {};
    for (int k = 0; k < K; k += 4) {
        v2f av;
        av.x = Arow[k + 2 * half];
        av.y = Arow[k + 2 * half + 1];
        const float* B0 = Bbase + (size_t)(k + 2 * half) * Ncols;
        const float* B1 = Bbase + (size_t)(k + 2 * half + 1) * Ncols;
        v2f b0, b1, b2, b3;
        b0.x = B0[0];  b0.y = B1[0];
        b1.x = B0[16]; b1.y = B1[16];
        b2.x = B0[32]; b2.y = B1[32];
        b3.x = B0[48]; b3.y = B1[48];
        acc0 = __builtin_amdgcn_wmma_f32_16x16x4_f32(false, av, false, b0, (short)0, acc0, false, false);
        acc1 = __builtin_amdgcn_wmma_f32_16x16x4_f32(false, av, false, b1, (short)0, acc1, false, false);
        acc2 = __builtin_amdgcn_wmma_f32_16x16x4_f32(false, av, false, b2, (short)0, acc2, false, false);
        acc3 = __builtin_amdgcn_wmma_f32_16x16x4_f32(false, av, false, b3, (short)0, acc3, false, false);
    }
    float* Crow = C + (size_t)(tm * 16 + half * 8) * Ncols + tn4 * 64 + r;
#pragma unroll
    for (int i = 0; i < 8; ++i) {
        size_t ro = (size_t)i * Ncols;
        Crow[ro + 0]  = acc0[i];
        Crow[ro + 16] = acc1[i];
        Crow[ro + 32] = acc2[i];
        Crow[ro + 48] = acc3[i];
    }
}

// K-split variant for the tall-skinny pooling GEMM (K=50000): partial slabs
// atomically combined into C (C must be zeroed first).
__global__ void gemm_wmma_f32_ksplit(const float* __restrict__ A,
                                     const float* __restrict__ B,
                                     float* __restrict__ C,
                                     int M, int K, int Ncols, int kChunk) {
    const int slabsN  = Ncols >> 6;
    const int slabsMN = (M >> 4) * slabsN;
    const int nSplit  = K / kChunk;
    int w = blockIdx.x * (blockDim.x >> 5) + (threadIdx.x >> 5);
    if (w >= slabsMN * nSplit) return;
    const int split = w / slabsMN;
    const int slab  = w % slabsMN;
    const int tm    = slab / slabsN;
    const int tn4   = slab % slabsN;
    const int lane  = threadIdx.x & 31;
    const int half  = lane >> 4;
    const int r     = lane & 15;

    const float* Arow  = A + (size_t)(tm * 16 + r) * K;
    const float* Bbase = B + tn4 * 64 + r;
    const int k0 = split * kChunk;

    v8f acc0 = {}, acc1 = {}, acc2 = {}, acc3 = {};
    for (int k = k0; k < k0 + kChunk; k += 4) {
        v2f av;
        av.x = Arow[k + 2 * half];
        av.y = Arow[k + 2 * half + 1];
        const float* B0 = Bbase + (size_t)(k + 2 * half) * Ncols;
        const float* B1 = Bbase + (size_t)(k + 2 * half + 1) * Ncols;
        v2f b0, b1, b2, b3;
        b0.x = B0[0];  b0.y = B1[0];
        b1.x = B0[16]; b1.y = B1[16];
        b2.x = B0[32]; b2.y = B1[32];
        b3.x = B0[48]; b3.y = B1[48];
        acc0 = __builtin_amdgcn_wmma_f32_16x16x4_f32(false, av, false, b0, (short)0, acc0, false, false);
        acc1 = __builtin_amdgcn_wmma_f32_16x16x4_f32(false, av, false, b1, (short)0, acc1, false, false);
        acc2 = __builtin_amdgcn_wmma_f32_16x16x4_f32(false, av, false, b2, (short)0, acc2, false, false);
        acc3 = __builtin_amdgcn_wmma_f32_16x16x4_f32(false, av, false, b3, (short)0, acc3, false, false);
    }
    float* Crow = C + (size_t)(tm * 16 + half * 8) * Ncols + tn4 * 64 + r;
#pragma unroll
    for (int i = 0; i < 8; ++i) {
        size_t ro = (size_t)i * Ncols;
        atomicAdd(&Crow[ro + 0],  acc0[i]);
        atomicAdd(&Crow[ro + 16], acc1[i]);
        atomicAdd(&Crow[ro + 32], acc2[i]);
        atomicAdd(&Crow[ro + 48], acc3[i]);
    }
}

// ---------------------------------------------------------------------------
// Per-node attention projections: e_src[n] = <xp[n,:], a_s>, e_dst likewise.
// One wave per node, shuffle reduction (wave32).
// ---------------------------------------------------------------------------
__global__ void attn_scores(const float* __restrict__ xp,
                            const float* __restrict__ a_s,
                            const float* __restrict__ a_d,
                            float* __restrict__ esrc,
                            float* __restrict__ edst, int C) {
    int node = blockIdx.x * (blockDim.x >> 5) + (threadIdx.x >> 5);
    if (node >= NN) return;
    int lane = threadIdx.x & 31;
    const float* xr = xp + (size_t)node * C;
    float s1 = 0.f, s2 = 0.f;
    for (int c = lane; c < C; c += 32) {
        float v = xr[c];
        s1 += v * a_s[c];
        s2 += v * a_d[c];
    }
    for (int off = 16; off; off >>= 1) {
        s1 += __shfl_xor(s1, off, 32);
        s2 += __shfl_xor(s2, off, 32);
    }
    if (lane == 0) { esrc[node] = s1; edst[node] = s2; }
}

// ---------------------------------------------------------------------------
// Edge pass 1: segment max of leaky_relu(e_src[s] + e_dst[d]) into amax[d].
// Float atomic max via int/uint punning (monotone bit patterns).
// ---------------------------------------------------------------------------
__device__ __forceinline__ void atomicMaxF(float* addr, float v) {
    if (v >= 0.f) atomicMax((int*)addr, __float_as_int(v));
    else          atomicMin((unsigned int*)addr, __float_as_uint(v));
}

__global__ void edge_max(const int* __restrict__ srcI, const int* __restrict__ dstI,
                         const float* __restrict__ esrc, const float* __restrict__ edst,
                         float* __restrict__ amax) {
    int e = blockIdx.x * blockDim.x + threadIdx.x;
    if (e >= ET) return;
    int s = srcI[e], d = dstI[e];
    float al = esrc[s] + edst[d];
    al = al > 0.f ? al : SLOPE * al;
    atomicMaxF(&amax[d], al);
}

// ---------------------------------------------------------------------------
// Edge pass 2: w = exp(al - amax[d]); den[d] += w; agg[d,:] += w * xp[s,:].
// One wave per edge. The 128/64-float source row is gathered as one b128
// per lane (rows are 256B aligned); scatter is native global_atomic_add_f32
// resolving in the 192MB L2 (feature matrix is only 25.6MB).
// ---------------------------------------------------------------------------
__global__ void edge_msg(const int* __restrict__ srcI, const int* __restrict__ dstI,
                         const float* __restrict__ esrc, const float* __restrict__ edst,
                         const float* __restrict__ amax, const float* __restrict__ xp,
                         float* __restrict__ den, float* __restrict__ agg, int C) {
    int e = blockIdx.x * (blockDim.x >> 5) + (threadIdx.x >> 5);
    if (e >= ET) return;
    int lane = threadIdx.x & 31;
    int s = srcI[e], d = dstI[e];
    float al = esrc[s] + edst[d];
    al = al > 0.f ? al : SLOPE * al;
    float w = __expf(al - amax[d]);
    if (lane == 0) atomicAdd(&den[d], w);
    const int nvec = C >> 2;                     // 32 (C=128) or 16 (C=64)
    if (lane < nvec) {
        const float4* xr = (const float4*)(xp + (size_t)s * C);
        float4 v = xr[lane];
        float* ar = agg + (size_t)d * C + lane * 4;
        atomicAdd(ar + 0, w * v.x);
        atomicAdd(ar + 1, w * v.y);
        atomicAdd(ar + 2, w * v.z);
        atomicAdd(ar + 3, w * v.w);
    }
}

// ---------------------------------------------------------------------------
// Finalize (agg/den + bias) in place and accumulate BN sums via LDS reduction.
// stats[0:C] = sum, stats[C:2C] = sumsq.
// ---------------------------------------------------------------------------
__global__ void finalize_stats(float* __restrict__ agg, const float* __restrict__ den,
                               const float* __restrict__ bias, float* __restrict__ stats,
                               int C) {
    __shared__ float ls[2 * DHC];
    for (int i = threadIdx.x; i < 2 * C; i += blockDim.x) ls[i] = 0.f;
    __syncthreads();
    const int perIter = blockDim.x / C;           // nodes handled per block iter
    const int c  = threadIdx.x % C;
    const int no = threadIdx.x / C;
    for (int n = blockIdx.x * perIter + no; n < NN; n += gridDim.x * perIter) {
        size_t idx = (size_t)n * C + c;
        float v = agg[idx] / den[n] + bias[c];
        agg[idx] = v;
        atomicAdd(&ls[c], v);
        atomicAdd(&ls[C + c], v * v);
    }
    __syncthreads();
    for (int i = threadIdx.x; i < 2 * C; i += blockDim.x) atomicAdd(&stats[i], ls[i]);
}

// stats[2C:3C] = mean, stats[3C:4C] = invstd  (biased variance, as in BN train)
__global__ void bn_reduce(float* __restrict__ stats, int C) {
    int c = threadIdx.x;
    if (c >= C) return;
    float mean = stats[c] / (float)NN;
    float var  = stats[C + c] / (float)NN - mean * mean;
    stats[2 * C + c] = mean;
    stats[3 * C + c] = rsqrtf(var + EPSV);
}

__global__ void bn_apply_relu(const float* __restrict__ h, const float* __restrict__ stats,
                              const float* __restrict__ g, const float* __restrict__ be,
                              float* __restrict__ out, int C) {
    size_t i = (size_t)blockIdx.x * blockDim.x + threadIdx.x;
    if (i >= (size_t)NN * C) return;
    int c = (int)(i & (C - 1));
    float v = (h[i] - stats[2 * C + c]) * stats[3 * C + c] * g[c] + be[c];
    out[i] = v > 0.f ? v : 0.f;
}

// Layer 3 finalize: h_nodes = agg/den + b3 (heads=1 -> mean is identity)
__global__ void finalize3(const float* __restrict__ agg, const float* __restrict__ den,
                          const float* __restrict__ bias, float* __restrict__ outNodes) {
    size_t i = (size_t)blockIdx.x * blockDim.x + threadIdx.x;
    if (i >= (size_t)NN * DOUTC) return;
    int c = (int)(i & (DOUTC - 1));
    int n = (int)(i >> 6);
    outNodes[i] = agg[i] / den[n] + bias[c];
}

// ---------------------------------------------------------------------------
// Host launcher
// ---------------------------------------------------------------------------
extern "C" void kernel_launch(void* const* d_in, const int* in_sizes, int n_in,
                              void* d_out, int out_size, void* d_ws, size_t ws_size,
                              hipStream_t stream) {
    const float*     x   = (const float*)d_in[0];
    const long long* ei  = (const long long*)d_in[1];
    const float*     gp  = (const float*)d_in[2];
    const float *W1 = (const float*)d_in[3],  *as1 = (const float*)d_in[4],
                *ad1 = (const float*)d_in[5], *b1  = (const float*)d_in[6],
                *g1  = (const float*)d_in[7], *be1 = (const float*)d_in[8];
    const float *W2 = (const float*)d_in[9],  *as2 = (const float*)d_in[10],
                *ad2 = (const float*)d_in[11],*b2  = (const float*)d_in[12],
                *g2  = (const float*)d_in[13],*be2 = (const float*)d_in[14];
    const float *W3 = (const float*)d_in[15], *as3 = (const float*)d_in[16],
                *ad3 = (const float*)d_in[17],*b3  = (const float*)d_in[18];

    // workspace layout
    float* fws  = (float*)d_ws;
    size_t nf   = (size_t)NN * DHC;
    float* bufA = fws;               // normalized activations (layer input)
    float* bufB = bufA + nf;         // xp = h @ W
    float* bufC = bufB + nf;         // aggregation / finalized output
    float* esrc = bufC + nf;
    float* edst = esrc + NN;
    float* amax = edst + NN;
    float* den  = amax + NN;
    float* stats = den + NN;         // 4*DHC
    int*   srcI  = (int*)(stats + 4 * DHC);
    int*   dstI  = srcI + ET;

    float* outPooled = (float*)d_out;                   // NGP x DOUTC
    float* outNodes  = outPooled + (size_t)NGP * DOUTC; // NN x DOUTC

    const float NEG_INF = -INFINITY;

    build_edges<<<(ET + 255) / 256, 256, 0, stream>>>(ei, srcI, dstI);

    // ---- per-layer helper macro (C output channels) ----
    #define GAT_LAYER(HIN, W, AS, AD, C)                                            \
        gemm_wmma_f32<<<((NN / 16) * ((C) / 64) + 3) / 4, 128, 0, stream>>>(        \
            (HIN), (W), bufB, NN, DHC, (C));                                        \
        attn_scores<<<(NN + 3) / 4, 128, 0, stream>>>(bufB, (AS), (AD), esrc, edst, (C)); \
        fill_f32<<<(NN + 255) / 256, 256, 0, stream>>>(amax, NN, NEG_INF);          \
        fill_f32<<<(NN + 255) / 256, 256, 0, stream>>>(den, NN, 0.f);               \
        fill_f32<<<((int)((size_t)NN * (C)) + 255) / 256, 256, 0, stream>>>(        \
            bufC, NN * (C), 0.f);                                                   \
        edge_max<<<(ET + 255) / 256, 256, 0, stream>>>(srcI, dstI, esrc, edst, amax); \
        edge_msg<<<(ET + 3) / 4, 128, 0, stream>>>(srcI, dstI, esrc, edst, amax,    \
                                                   bufB, den, bufC, (C));

    // ===== layer 1 (DIN=128 -> DHC=128) =====
    GAT_LAYER(x, W1, as1, ad1, DHC)
    fill_f32<<<(4 * DHC + 255) / 256, 256, 0, stream>>>(stats, 4 * DHC, 0.f);
    finalize_stats<<<784, 256, 0, stream>>>(bufC, den, b1, stats, DHC);
    bn_reduce<<<1, DHC, 0, stream>>>(stats, DHC);
    bn_apply_relu<<<(int)((nf + 255) / 256), 256, 0, stream>>>(bufC, stats, g1, be1, bufA, DHC);

    // ===== layer 2 (DHC -> DHC) =====
    GAT_LAYER(bufA, W2, as2, ad2, DHC)
    fill_f32<<<(4 * DHC + 255) / 256, 256, 0, stream>>>(stats, 4 * DHC, 0.f);
    finalize_stats<<<784, 256, 0, stream>>>(bufC, den, b2, stats, DHC);
    bn_reduce<<<1, DHC, 0, stream>>>(stats, DHC);
    bn_apply_relu<<<(int)((nf + 255) / 256), 256, 0, stream>>>(bufC, stats, g2, be2, bufA, DHC);

    // ===== layer 3 (DHC -> DOUTC, no BN/ReLU) =====
    GAT_LAYER(bufA, W3, as3, ad3, DOUTC)
    finalize3<<<(int)(((size_t)NN * DOUTC + 255) / 256), 256, 0, stream>>>(
        bufC, den, b3, outNodes);

    // ===== pooling: h_pooled = graph_pool (128 x 50000) @ h_nodes (50000 x 64) =====
    fill_f32<<<(NGP * DOUTC + 255) / 256, 256, 0, stream>>>(outPooled, NGP * DOUTC, 0.f);
    {
        const int kChunk = 1000;                       // 50 K-splits
        const int waves  = (NGP / 16) * (DOUTC / 64) * (NN / kChunk);
        gemm_wmma_f32_ksplit<<<(waves + 3) / 4, 128, 0, stream>>>(
            gp, outNodes, outPooled, NGP, NN, DOUTC, kChunk);
    }
    #undef GAT_LAYER
}